// CDEPdeLayerR2_10101763080236
// MI455X (gfx1250) — compile-verified
//
#include <hip/hip_runtime.h>
#include <math.h>

typedef float v2f __attribute__((ext_vector_type(2)));
typedef float v8f __attribute__((ext_vector_type(8)));

#define HH 256
#define WW 256
#define CIN 32
#define COUT 32
#define BATCH 16
#define RAD 3
#define KDIA 7
#define TILE 32
#define S1D 44   // TILE + 4*RAD
#define S2D 38   // TILE + 2*RAD
#define BIGF 1e30f

// ---------------------------------------------------------------------------
// Kernel 0: sample the fractional morphological kernels (32 ch x 7 x 7, x2)
// ---------------------------------------------------------------------------
__global__ void make_kernels(const float* __restrict__ fd,
                             const float* __restrict__ fe,
                             float* __restrict__ kd,
                             float* __restrict__ ke) {
    int c = blockIdx.x;
    int t = threadIdx.x;
    if (t >= 49) return;
    float dy = (float)(t / 7) - 3.0f;
    float dx = (float)(t % 7) - 3.0f;
    float rho = sqrtf(dx * dx + dy * dy);
    float th  = atan2f(dy, dx);
    float b0 = cosf(th), b1 = sinf(th);
    float b2 = cosf(2.0f * th), b3 = sinf(2.0f * th);
    const float p  = 2.0f * 0.65f / (2.0f * 0.65f - 1.0f);     // 13/3
    const float nu = (2.0f * 0.65f - 1.0f) * powf(2.0f * 0.65f, -p);
    {
        const float* f = fd + c * 4;
        float F = rho * expf(-(f[0] * b0 + f[1] * b1 + f[2] * b2 + f[3] * b3));
        kd[c * 49 + t] = nu * powf(F, p);                      // pow(0,p)=0 at center
    }
    {
        const float* f = fe + c * 4;
        float F = rho * expf(-(f[0] * b0 + f[1] * b1 + f[2] * b2 + f[3] * b3));
        ke[c * 49 + t] = nu * powf(F, p);
    }
}

// ---------------------------------------------------------------------------
// Kernel 1: fused convection (bilinear per-channel shift) -> dilation ->
//           erosion, tiled through LDS.  One WG per (b*c, 32x32 tile).
// ---------------------------------------------------------------------------
__global__ void __launch_bounds__(256)
fused_morph(const float* __restrict__ x,
            const float* __restrict__ c,
            const float* __restrict__ kd,
            const float* __restrict__ ke,
            float* __restrict__ u) {
    __shared__ float s1[S1D * S1D];   // convected tile (+6 halo), -BIG outside image
    __shared__ float s2[S2D * S2D];   // dilated tile (+3 halo),  +BIG outside image
    __shared__ float lkd[49];
    __shared__ float lke[49];

    int bc  = blockIdx.z;             // b*CIN + ch
    int ch  = bc & (CIN - 1);
    int ty0 = blockIdx.y * TILE;
    int tx0 = blockIdx.x * TILE;
    int tid = threadIdx.x;

    if (tid < 49) { lkd[tid] = kd[ch * 49 + tid]; lke[tid] = ke[ch * 49 + tid]; }
    float cx = c[ch * 2 + 0];
    float cy = c[ch * 2 + 1];
    const float* xc = x + (size_t)bc * (HH * WW);

    // Phase A: convection into s1
    for (int i = tid; i < S1D * S1D; i += 256) {
        int ly = i / S1D, lx = i % S1D;
        int gy = ty0 + ly - 2 * RAD;
        int gx = tx0 + lx - 2 * RAD;
        float v;
        if (gy >= 0 && gy < HH && gx >= 0 && gx < WW) {
            float fy = (float)gy - cy;
            float fx = (float)gx - cx;
            float y0f = floorf(fy), x0f = floorf(fx);
            float wy = fy - y0f, wx = fx - x0f;
            int y0 = min(max((int)y0f, 0), HH - 1);
            int x0 = min(max((int)x0f, 0), WW - 1);
            int y1 = min(y0 + 1, HH - 1);
            int x1 = min(x0 + 1, WW - 1);
            float v00 = xc[y0 * WW + x0];
            float v01 = xc[y0 * WW + x1];
            float v10 = xc[y1 * WW + x0];
            float v11 = xc[y1 * WW + x1];
            v = (1.f - wy) * (1.f - wx) * v00 + (1.f - wy) * wx * v01
              + wy * (1.f - wx) * v10 + wy * wx * v11;
        } else {
            v = -BIGF;                 // excluded from the max (dilation)
        }
        s1[i] = v;
    }
    __syncthreads();

    // Phase B: dilation  d = max_taps (u - k_dil)
    for (int i = tid; i < S2D * S2D; i += 256) {
        int ly = i / S2D, lx = i % S2D;
        int gy = ty0 + ly - RAD;
        int gx = tx0 + lx - RAD;
        float m;
        if (gy >= 0 && gy < HH && gx >= 0 && gx < WW) {
            m = -BIGF;
            #pragma unroll
            for (int dy = 0; dy < KDIA; ++dy)
                #pragma unroll
                for (int dx = 0; dx < KDIA; ++dx)
                    m = fmaxf(m, s1[(ly + dy) * S1D + (lx + dx)] - lkd[dy * KDIA + dx]);
        } else {
            m = BIGF;                  // excluded from the min (erosion)
        }
        s2[i] = m;
    }
    __syncthreads();

    // Phase C: erosion  e = min_taps (d + k_ero), store u
    float* uo = u + (size_t)bc * (HH * WW);
    for (int i = tid; i < TILE * TILE; i += 256) {
        int ly = i / TILE, lx = i % TILE;
        float m = BIGF;
        #pragma unroll
        for (int dy = 0; dy < KDIA; ++dy)
            #pragma unroll
            for (int dx = 0; dx < KDIA; ++dx)
                m = fminf(m, s2[(ly + dy) * S2D + (lx + dx)] + lke[dy * KDIA + dx]);
        uo[(ty0 + ly) * WW + (tx0 + lx)] = m;
    }
}

// ---------------------------------------------------------------------------
// Kernel 2: channel mix  out[b,o,p] = sum_k W[k,o] * u[b,k,p]
// full-fp32 WMMA 16x16x4: D[o, pixel] tiles, M=o (2 tiles of 16), N=16 pixels,
// K=Cin=32 as 8 chained x4 steps.  One wave per 16 pixels.
// ---------------------------------------------------------------------------
__global__ void __launch_bounds__(256)
mix_wmma(const float* __restrict__ u,
         const float* __restrict__ wgt,
         float* __restrict__ out) {
    int lane = threadIdx.x & 31;
    int wv   = (blockIdx.x * 256 + threadIdx.x) >> 5;   // global wave id
    int half = lane >> 4;                               // 0: lanes 0-15, 1: 16-31
    int l15  = lane & 15;

    size_t pixbase = (size_t)wv * 16;                   // flat pixel over (b,h,w)
    int b    = (int)(pixbase >> 16);                    // 65536 pixels per image
    int pinb = (int)(pixbase & 65535);

    const float* ub = u   + ((size_t)b * CIN  << 16);
    float*       ob = out + ((size_t)b * COUT << 16);

    v8f acc0 = {0.f, 0.f, 0.f, 0.f, 0.f, 0.f, 0.f, 0.f};
    v8f acc1 = {0.f, 0.f, 0.f, 0.f, 0.f, 0.f, 0.f, 0.f};

    #pragma unroll
    for (int s = 0; s < 8; ++s) {
        // A 16x4 fp32 layout: v0 holds K=k0 (lanes 0-15) / K=k0+2 (lanes 16-31),
        // v1 holds K=k0+1 / K=k0+3; M = lane&15.  A = W^T (A[m=o,k] = W[k,o]).
        int k0 = 4 * s + half * 2;
        v2f a0, a1, bf;
        a0[0] = wgt[k0 * COUT + l15];
        a0[1] = wgt[(k0 + 1) * COUT + l15];
        a1[0] = wgt[k0 * COUT + 16 + l15];
        a1[1] = wgt[(k0 + 1) * COUT + 16 + l15];
        // B 4x16: row K striped across lanes within a VGPR, same half-split.
        bf[0] = ub[((size_t)k0 << 16) + pinb + l15];
        bf[1] = ub[((size_t)(k0 + 1) << 16) + pinb + l15];

        acc0 = __builtin_amdgcn_wmma_f32_16x16x4_f32(false, a0, false, bf,
                                                     (short)0, acc0, false, false);
        acc1 = __builtin_amdgcn_wmma_f32_16x16x4_f32(false, a1, false, bf,
                                                     (short)0, acc1, false, false);
    }

    // D 16x16 fp32: VGPR j holds M=j (lanes 0-15) / M=j+8 (lanes 16-31), N=lane&15
    #pragma unroll
    for (int j = 0; j < 8; ++j) {
        int o = j + half * 8;
        ob[((size_t)o << 16) + pinb + l15]        = acc0[j];
        ob[((size_t)(o + 16) << 16) + pinb + l15] = acc1[j];
    }
}

// ---------------------------------------------------------------------------
extern "C" void kernel_launch(void* const* d_in, const int* in_sizes, int n_in,
                              void* d_out, int out_size, void* d_ws, size_t ws_size,
                              hipStream_t stream) {
    const float* x   = (const float*)d_in[0];   // (16,32,256,256)
    const float* c   = (const float*)d_in[1];   // (32,2)
    const float* fd  = (const float*)d_in[2];   // (32,4)
    const float* fe  = (const float*)d_in[3];   // (32,4)
    const float* wgt = (const float*)d_in[4];   // (32,32)
    float* out = (float*)d_out;                 // (16,32,256,256)

    char*  ws = (char*)d_ws;
    float* kd = (float*)ws;                               // 32*49 floats
    float* ke = (float*)(ws + CIN * 49 * sizeof(float));  // 32*49 floats
    float* u  = (float*)(ws + 16384);                     // (16,32,256,256) = 128 MiB

    make_kernels<<<dim3(CIN), dim3(64), 0, stream>>>(fd, fe, kd, ke);

    fused_morph<<<dim3(WW / TILE, HH / TILE, BATCH * CIN), dim3(256), 0, stream>>>(
        x, c, kd, ke, u);

    int total_waves = BATCH * HH * WW / 16;   // 65536 waves, 16 pixels each
    mix_wmma<<<dim3(total_waves / 8), dim3(256), 0, stream>>>(u, wgt, out);
}